// SpatialMaxpool2d_2_58772332478720
// MI455X (gfx1250) — compile-verified
//
#include <hip/hip_runtime.h>
#include <hip/hip_bf16.h>

// ---------------------------------------------------------------------------
// CDNA5 async global->LDS helpers (gfx1250).
// ---------------------------------------------------------------------------
typedef int v4i __attribute__((vector_size(16)));
typedef __attribute__((address_space(1))) v4i* gptr_v4i;
typedef __attribute__((address_space(3))) v4i* lptr_v4i;

__device__ __forceinline__ void async_load_b128(const float* g, float* l) {
#if __has_builtin(__builtin_amdgcn_global_load_async_to_lds_b128)
    __builtin_amdgcn_global_load_async_to_lds_b128(
        (gptr_v4i)(v4i*)g, (lptr_v4i)(v4i*)l, 0, 0);
#else
    unsigned loff = (unsigned)(unsigned long long)
        (__attribute__((address_space(3))) void*)l;
    asm volatile("global_load_async_to_lds_b128 %0, %1, off"
                 :: "v"(loff), "v"(g) : "memory");
#endif
}

template <int N>
__device__ __forceinline__ void wait_async() {
#if __has_builtin(__builtin_amdgcn_s_wait_asynccnt)
    __builtin_amdgcn_s_wait_asynccnt(N);
#else
    asm volatile("s_wait_asynccnt %0" :: "i"(N) : "memory");
#endif
}

// Wait until async loads for channel `c` have landed; `rem` = channels still
// outstanding after c (2 async instructions per channel, in-order completion).
__device__ __forceinline__ void wait_for_rem(int rem) {
    if      (rem >= 7) wait_async<14>();
    else if (rem == 6) wait_async<12>();
    else if (rem == 5) wait_async<10>();
    else if (rem == 4) wait_async<8>();
    else if (rem == 3) wait_async<6>();
    else if (rem == 2) wait_async<4>();
    else if (rem == 1) wait_async<2>();
    else               wait_async<0>();
}

// ---------------------------------------------------------------------------
// Kernel 1: maxpool (channels 0..255) + shape_weights scatter-accumulate.
//   grid = (8 row-groups, 16 batches), block = 256 (8 waves).
//   Each wave privately owns ONE pooled row (input rows 2hp, 2hp+1) for all
//   256 channels: lane L async-loads 16B of each row, consumes only its own
//   wave's data -> NO workgroup barriers, per-wave ASYNCcnt is the only sync.
//   Lane L handles pooled cols 2L, 2L+1; argmax-scatter accumulates in regs.
// ---------------------------------------------------------------------------
#define NBUF 8   // channel-deep software pipeline (2 async b128 per channel)

__global__ __launch_bounds__(256) void k_pool_sw(const float* __restrict__ x,
                                                 float* __restrict__ out,
                                                 float* __restrict__ sw) {
    __shared__ float buf[8][NBUF][256];   // 8 waves x 8 bufs x (2 rows x 128) = 64 KB

    const int b    = blockIdx.y;              // 0..15 batch
    const int tid  = threadIdx.x;
    const int w    = tid >> 5;                // wave id 0..7
    const int lane = tid & 31;
    const int hp   = blockIdx.x * 8 + w;      // pooled row 0..63

    const size_t plane = 128 * 128;
    // global base of this wave's row pair for channel 0
    const float* xr = x + (size_t)b * 261 * plane + (size_t)(2 * hp) * 128;
    const int gl = 4 * lane;                  // lane's float offset within a row

    // prologue: fill the pipeline (channels 0..NBUF-1)
    #pragma unroll
    for (int c = 0; c < NBUF; ++c) {
        const float* xc = xr + (size_t)c * plane;
        async_load_b128(xc + gl,        &buf[w][c][gl]);
        async_load_b128(xc + 128 + gl,  &buf[w][c][128 + gl]);
    }

    float a00 = 0.f, a01 = 0.f, a02 = 0.f, a03 = 0.f;   // window 0 (cols 4L,4L+1)
    float a10 = 0.f, a11 = 0.f, a12 = 0.f, a13 = 0.f;   // window 1 (cols 4L+2,4L+3)

    float* outp = out + (size_t)b * 261 * 4096 + (size_t)hp * 64 + 2 * lane;

    auto consume = [&](int c) {
        const float* bp = &buf[w][c & (NBUF - 1)][0];
        const float4 tr = *(const float4*)&bp[gl];         // row 2hp, cols 4L..4L+3
        const float4 br = *(const float4*)&bp[128 + gl];   // row 2hp+1

        // argmax order (kh,kw): (0,0),(0,1),(1,0),(1,1); first-max wins
        float m0 = tr.x; int p0 = 0;
        if (tr.y > m0) { m0 = tr.y; p0 = 1; }
        if (br.x > m0) { m0 = br.x; p0 = 2; }
        if (br.y > m0) { m0 = br.y; p0 = 3; }

        float m1 = tr.z; int p1 = 0;
        if (tr.w > m1) { m1 = tr.w; p1 = 1; }
        if (br.z > m1) { m1 = br.z; p1 = 2; }
        if (br.w > m1) { m1 = br.w; p1 = 3; }

        *(float2*)(outp + (size_t)c * 4096) = make_float2(m0, m1);

        const float am0 = fabsf(m0), am1 = fabsf(m1);
        if      (p0 == 0) a00 += am0;
        else if (p0 == 1) a01 += am0;
        else if (p0 == 2) a02 += am0;
        else              a03 += am0;
        if      (p1 == 0) a10 += am1;
        else if (p1 == 1) a11 += am1;
        else if (p1 == 2) a12 += am1;
        else              a13 += am1;
    };

    // steady state: wait oldest channel, consume it, refill its buffer
    for (int c = 0; c < 256 - NBUF; ++c) {
        wait_for_rem(7);                      // keep NBUF-1 channels in flight
        consume(c);
        const int nc = c + NBUF;
        const float* xc = xr + (size_t)nc * plane;
        float* lb = &buf[w][nc & (NBUF - 1)][0];
        async_load_b128(xc + gl,       lb + gl);
        async_load_b128(xc + 128 + gl, lb + 128 + gl);
    }
    // drain tail: rem is a compile-time constant per unrolled iteration
    #pragma unroll
    for (int k = 0; k < NBUF; ++k) {
        const int c = 256 - NBUF + k;
        wait_for_rem(255 - c);
        consume(c);
    }

    // shape_weights = (sum_c |unpooled|) / sqrt(256); lane owns cols 4L..4L+3
    float* swp = sw + (size_t)b * plane + (size_t)(2 * hp) * 128 + gl;
    *(float4*)swp         = make_float4(a00 * 0.0625f, a01 * 0.0625f,
                                        a10 * 0.0625f, a11 * 0.0625f);
    *(float4*)(swp + 128) = make_float4(a02 * 0.0625f, a03 * 0.0625f,
                                        a12 * 0.0625f, a13 * 0.0625f);
}

// ---------------------------------------------------------------------------
// Kernel 2: per-window weighted stats (output channels 256..260).
//   grid = (16, 16 batches), block = 256; one thread per output pixel.
// ---------------------------------------------------------------------------
__global__ __launch_bounds__(256) void k_stats(const float* __restrict__ x,
                                               const float* __restrict__ sw,
                                               float* __restrict__ out) {
    const int b   = blockIdx.y;
    const int idx = blockIdx.x * 256 + threadIdx.x;  // 0..4095
    const int ii  = idx >> 6;
    const int jj  = idx & 63;

    const size_t plane = 128 * 128;
    const size_t pos   = (size_t)(2 * ii) * 128 + 2 * jj;

    const float* swp = sw + (size_t)b * plane + pos;
    const float2 qt = *(const float2*)swp;
    const float2 qb = *(const float2*)(swp + 128);
    float q0 = qt.x, q1 = qt.y, q2 = qb.x, q3 = qb.y;
    const float s = fabsf(q0) + fabsf(q1) + fabsf(q2) + fabsf(q3) + 0.001f;
    q0 /= s; q1 /= s; q2 /= s; q3 /= s;

    const float* xb = x + (size_t)b * 261 * plane + pos;
    const float2 c0t = *(const float2*)(xb + 256 * plane);
    const float2 c0b = *(const float2*)(xb + 256 * plane + 128);
    const float2 c1t = *(const float2*)(xb + 257 * plane);
    const float2 c1b = *(const float2*)(xb + 257 * plane + 128);
    const float2 v0t = *(const float2*)(xb + 258 * plane);
    const float2 v0b = *(const float2*)(xb + 258 * plane + 128);
    const float2 v1t = *(const float2*)(xb + 259 * plane);
    const float2 v1b = *(const float2*)(xb + 259 * plane + 128);
    const float2 cvt = *(const float2*)(xb + 260 * plane);
    const float2 cvb = *(const float2*)(xb + 260 * plane + 128);

    const float mean0 = c0t.x * q0 + c0t.y * q1 + c0b.x * q2 + c0b.y * q3;
    const float mean1 = c1t.x * q0 + c1t.y * q1 + c1b.x * q2 + c1b.y * q3;

    float var0 = v0t.x * q0 + v0t.y * q1 + v0b.x * q2 + v0b.y * q3;
    float var1 = v1t.x * q0 + v1t.y * q1 + v1b.x * q2 + v1b.y * q3;

    const float d00 = c0t.x - mean0, d01 = c0t.y - mean0,
                d02 = c0b.x - mean0, d03 = c0b.y - mean0;
    const float d10 = c1t.x - mean1, d11 = c1t.y - mean1,
                d12 = c1b.x - mean1, d13 = c1b.y - mean1;

    var0 += d00 * d00 * q0 + d01 * d01 * q1 + d02 * d02 * q2 + d03 * d03 * q3;
    var1 += d10 * d10 * q0 + d11 * d11 * q1 + d12 * d12 * q2 + d13 * d13 * q3;

    const float cov = cvt.x * q0 + cvt.y * q1 + cvb.x * q2 + cvb.y * q3
                    + d00 * d10 * q0 + d01 * d11 * q1 + d02 * d12 * q2 + d03 * d13 * q3;

    float* ob = out + (size_t)b * 261 * 4096 + (size_t)ii * 64 + jj;
    ob[(size_t)256 * 4096] = mean0;
    ob[(size_t)257 * 4096] = mean1;
    ob[(size_t)258 * 4096] = var0;
    ob[(size_t)259 * 4096] = var1;
    ob[(size_t)260 * 4096] = cov;
}

// ---------------------------------------------------------------------------
extern "C" void kernel_launch(void* const* d_in, const int* in_sizes, int n_in,
                              void* d_out, int out_size, void* d_ws, size_t ws_size,
                              hipStream_t stream) {
    const float* x  = (const float*)d_in[0];  // [16, 261, 128, 128] f32
    float* out      = (float*)d_out;          // [16, 261, 64, 64] f32
    float* sw       = (float*)d_ws;           // [16, 128, 128] f32 scratch (1 MB)

    dim3 blk(256);
    dim3 g1(8, 16);
    k_pool_sw<<<g1, blk, 0, stream>>>(x, out, sw);

    dim3 g2(16, 16);
    k_stats<<<g2, blk, 0, stream>>>(x, sw, out);
}